// SingleGCN_26190710571456
// MI455X (gfx1250) — compile-verified
//
#include <hip/hip_runtime.h>
#include <stdint.h>

#define NN 100000            // total nodes (60000 + 40000)
#define EE 1200000           // edges
#define DD 64                // feature dim (in == out)
#define NTILES (NN / 16)     // 6250 exact 16-row tiles
#define HTILES (NTILES / 2)  // 3125: wave handles tile t and t+HTILES
#define HALF   ((uint32_t)(NN * DD) >> 1)  // 3,200,000 = 50000*64

typedef float v2f __attribute__((ext_vector_type(2)));
typedef float v8f __attribute__((ext_vector_type(8)));

// ---------------------------------------------------------------------------
// JAX threefry2x32 (20 rounds). key = jax.random.key(42) -> (k0,k1) = (0,42)
// ---------------------------------------------------------------------------
__device__ __forceinline__ uint2 threefry2x32(uint32_t k0, uint32_t k1,
                                              uint32_t c0, uint32_t c1) {
  uint32_t ks2 = k0 ^ k1 ^ 0x1BD11BDAu;
  uint32_t x0 = c0 + k0, x1 = c1 + k1;
#define TF_ROUND(r) { x0 += x1; x1 = (x1 << (r)) | (x1 >> (32 - (r))); x1 ^= x0; }
  TF_ROUND(13) TF_ROUND(15) TF_ROUND(26) TF_ROUND(6)
  x0 += k1;  x1 += ks2 + 1u;
  TF_ROUND(17) TF_ROUND(29) TF_ROUND(16) TF_ROUND(24)
  x0 += ks2; x1 += k0 + 2u;
  TF_ROUND(13) TF_ROUND(15) TF_ROUND(26) TF_ROUND(6)
  x0 += k0;  x1 += k1 + 3u;
  TF_ROUND(17) TF_ROUND(29) TF_ROUND(16) TF_ROUND(24)
  x0 += k1;  x1 += ks2 + 4u;
  TF_ROUND(13) TF_ROUND(15) TF_ROUND(26) TF_ROUND(6)
  x0 += ks2; x1 += k0 + 5u;
#undef TF_ROUND
  return make_uint2(x0, x1);
}

// bits -> uniform [0,1) < keep ?
__device__ __forceinline__ bool keep_bit(uint32_t bits) {
  float u = __uint_as_float((bits >> 9) | 0x3F800000u) - 1.0f;
  return u < 0.6f;
}

// ---------------------------------------------------------------------------
// 1) zero deg[] and out[] (float4)
// ---------------------------------------------------------------------------
__global__ void zero_kernel(float4* __restrict__ deg4, float4* __restrict__ out4) {
  uint32_t i = blockIdx.x * blockDim.x + threadIdx.x;
  const float4 z = make_float4(0.f, 0.f, 0.f, 0.f);
  if (i < (NN / 4)) deg4[i] = z;
  else if (i < (uint32_t)((NN + NN * DD) / 4)) out4[i - NN / 4] = z;
}

// ---------------------------------------------------------------------------
// 2) deg[col[e]] += w[e]
// ---------------------------------------------------------------------------
__global__ void deg_kernel(const int* __restrict__ col, const float* __restrict__ w,
                           float* __restrict__ deg) {
  uint32_t e = blockIdx.x * blockDim.x + threadIdx.x;
  if (e >= EE) return;
  unsafeAtomicAdd(&deg[col[e]], w[e]);
}

// ---------------------------------------------------------------------------
// 3) dinv = deg > 0 ? rsqrt(deg) : 0
// ---------------------------------------------------------------------------
__global__ void dinv_kernel(const float* __restrict__ deg, float* __restrict__ dinv) {
  uint32_t i = blockIdx.x * blockDim.x + threadIdx.x;
  if (i >= NN) return;
  float d = deg[i];
  dinv[i] = (d > 0.0f) ? rsqrtf(d) : 0.0f;
}

// ---------------------------------------------------------------------------
// 4) h = dropout(x) @ W via V_WMMA_F32_16X16X4_F32.
// Each wave owns TWO 16-row tiles: rows [r0,r0+16) and [r0+50000, ...).
// Flat indices differ by exactly HALF, so ONE threefry eval produces the
// dropout bits for both tiles (o.x / o.y).
// Loop order: k outer, nt inner — all 8 accumulators (64 VGPRs) stay live,
// A fragments are produced per k-step and consumed immediately by 8 WMMAs.
// This caps the live set (~90 VGPRs) and avoids the scratch spills seen when
// precomputing 32 A fragments up front.
// ---------------------------------------------------------------------------
__global__ void __launch_bounds__(256)
gemm_dropout_wmma(const float* __restrict__ x,
                  const float* __restrict__ W,
                  float* __restrict__ h) {
  __shared__ float Wl[DD * DD];
  // stage W (16 KB) into LDS; barrier BEFORE the wave-uniform tile guard
  {
    const float4* Wv = (const float4*)W;
    float4* Wlv = (float4*)Wl;
#pragma unroll
    for (int i = threadIdx.x; i < DD * DD / 4; i += 256) Wlv[i] = Wv[i];
  }
  __syncthreads();

  const int lane = threadIdx.x & 31;
  const int wave = threadIdx.x >> 5;
  const int tile = blockIdx.x * 8 + wave;
  if (tile >= HTILES) return;                 // wave-uniform

  const int row0 = tile * 16;                 // lower tile
  const int row1 = row0 + NN / 2;             // partner tile (+50000 rows)
  const int m    = lane & 15;
  const int khi  = (lane >> 4) << 1;          // 0 | 2
  const uint32_t base0 = (uint32_t)(row0 + m) * DD;

  v8f c[8];                                   // [nt*2 + tile] accumulators
#pragma unroll
  for (int i = 0; i < 8; ++i) c[i] = (v8f){0.f,0.f,0.f,0.f,0.f,0.f,0.f,0.f};

#pragma unroll
  for (int k = 0; k < 16; ++k) {
    const int k4 = k * 4;
    // ---- A fragments for this k-step (both row-tiles), dropout fused ----
    uint32_t i0 = base0 + (uint32_t)(k4 + khi);          // even -> 8B aligned
    float2 xl = *(const float2*)(x + i0);                // rows r0+m
    float2 xh = *(const float2*)(x + i0 + HALF);         // rows r1+m
    uint2 oA = threefry2x32(0u, 42u, i0,     i0 + HALF);
    uint2 oB = threefry2x32(0u, 42u, i0 + 1, i0 + 1 + HALF);
    const float inv = 1.0f / 0.6f;
    v2f a0, a1;
    a0.x = keep_bit(oA.x) ? xl.x * inv : 0.0f;
    a0.y = keep_bit(oB.x) ? xl.y * inv : 0.0f;
    a1.x = keep_bit(oA.y) ? xh.x * inv : 0.0f;
    a1.y = keep_bit(oB.y) ? xh.y * inv : 0.0f;

    // ---- 4 N-tiles x 2 row-tiles: 8 WMMAs sharing this A pair ----
#pragma unroll
    for (int nt = 0; nt < 4; ++nt) {
      const int n0 = nt * 16;
      v2f b;                                   // B 4x16 fragment from LDS
      b.x = Wl[(k4 + khi) * DD + n0 + m];
      b.y = Wl[(k4 + khi + 1) * DD + n0 + m];
      c[nt * 2 + 0] = __builtin_amdgcn_wmma_f32_16x16x4_f32(
          false, a0, false, b, (short)0, c[nt * 2 + 0], false, false);
      c[nt * 2 + 1] = __builtin_amdgcn_wmma_f32_16x16x4_f32(
          false, a1, false, b, (short)0, c[nt * 2 + 1], false, false);
    }
  }

  // ---- store D tiles: VGPR v -> row v (lanes 0-15) | v+8 (lanes 16-31) ----
#pragma unroll
  for (int nt = 0; nt < 4; ++nt) {
    const int n0 = nt * 16;
#pragma unroll
    for (int v = 0; v < 8; ++v) {
      int dr = (lane < 16) ? v : v + 8;
      h[(size_t)(row0 + dr) * DD + n0 + m] = c[nt * 2 + 0][v];
      h[(size_t)(row1 + dr) * DD + n0 + m] = c[nt * 2 + 1][v];
    }
  }
}

// ---------------------------------------------------------------------------
// 5) out[col] += (dinv[row]*w*dinv[col]) * h[row]; one wave per edge.
// Edge index forced scalar (readfirstlane) so row/col/w come from s_load.
// ---------------------------------------------------------------------------
__global__ void scatter_kernel(const int* __restrict__ row, const int* __restrict__ col,
                               const float* __restrict__ w, const float* __restrict__ dinv,
                               const float* __restrict__ h, float* __restrict__ out) {
  uint32_t wid  = (uint32_t)__builtin_amdgcn_readfirstlane(threadIdx.x >> 5);
  uint32_t e    = blockIdx.x * 8u + wid;        // wave-uniform in an SGPR
  uint32_t lane = threadIdx.x & 31;
  if (e >= EE) return;
  int r = row[e], c = col[e];
  float norm = dinv[r] * w[e] * dinv[c];
  const float2 hv = ((const float2*)(h + (size_t)r * DD))[lane];
  float* dst = out + (size_t)c * DD + lane * 2;
  unsafeAtomicAdd(dst,     norm * hv.x);
  unsafeAtomicAdd(dst + 1, norm * hv.y);
}

// ---------------------------------------------------------------------------
// 6) in-place ReLU (float4)
// ---------------------------------------------------------------------------
__global__ void relu_kernel(float4* __restrict__ out4) {
  uint32_t i = blockIdx.x * blockDim.x + threadIdx.x;
  if (i >= (uint32_t)(NN * DD / 4)) return;
  float4 v = out4[i];
  v.x = fmaxf(v.x, 0.f); v.y = fmaxf(v.y, 0.f);
  v.z = fmaxf(v.z, 0.f); v.w = fmaxf(v.w, 0.f);
  out4[i] = v;
}

// ---------------------------------------------------------------------------
extern "C" void kernel_launch(void* const* d_in, const int* in_sizes, int n_in,
                              void* d_out, int out_size, void* d_ws, size_t ws_size,
                              hipStream_t stream) {
  const float* x  = (const float*)d_in[0];          // [N, 64]
  const int*   ei = (const int*)d_in[1];            // [2, E] row-major
  const float* ew = (const float*)d_in[2];          // [E]
  const float* W  = (const float*)d_in[3];          // [64, 64]
  float* out = (float*)d_out;                       // [N, 64]

  float* ws   = (float*)d_ws;
  float* deg  = ws;                                 // N floats
  float* dinv = ws + NN;                            // N floats
  float* h    = ws + 2 * (size_t)NN;                // N*64 floats

  const int* rowp = ei;                             // edge_index[0]
  const int* colp = ei + EE;                        // edge_index[1]

  const int B = 256;

  // 1) zero deg + out accumulators
  {
    uint32_t total4 = (NN + NN * DD) / 4;
    zero_kernel<<<(total4 + B - 1) / B, B, 0, stream>>>((float4*)deg, (float4*)out);
  }
  // 2) degree
  deg_kernel<<<(EE + B - 1) / B, B, 0, stream>>>(colp, ew, deg);
  // 3) dinv
  dinv_kernel<<<(NN + B - 1) / B, B, 0, stream>>>(deg, dinv);
  // 4) h = dropout(x) @ W : one wave per PAIR of 16-row tiles
  {
    int blocks = (HTILES + 7) / 8;                  // 8 waves/block
    gemm_dropout_wmma<<<blocks, B, 0, stream>>>(x, W, h);
  }
  // 5) SpMM scatter: one wave per edge
  {
    uint64_t threads = (uint64_t)EE * 32;
    scatter_kernel<<<(uint32_t)((threads + B - 1) / B), B, 0, stream>>>(
        rowp, colp, ew, dinv, h, out);
  }
  // 6) ReLU
  relu_kernel<<<(NN * DD / 4 + B - 1) / B, B, 0, stream>>>((float4*)out);
}